// GraphSAGE_9766755631343
// MI455X (gfx1250) — compile-verified
//
#include <hip/hip_runtime.h>
#include <hip/hip_bf16.h>

// GraphSAGE 3-layer forward for MI455X (gfx1250).
// Memory-bound on edge gather/scatter-add; GEMMs use fp32 WMMA
// (v_wmma_f32_16x16x4_f32) so precision matches the fp32 reference.
// Each GEMM wave owns a full 16x128 output row band: A fragments are loaded
// once and reused across all 8 column tiles (8 v8f accumulators).

#define NNODES 100000
#define FDIM   128

typedef __attribute__((ext_vector_type(2))) float v2f;
typedef __attribute__((ext_vector_type(8))) float v8f;

// ---------------------------------------------------------------- utilities

__global__ void sage_zero_f32(float* __restrict__ p, long long n) {
    long long i = (long long)blockIdx.x * blockDim.x + threadIdx.x;
    if (i < n) p[i] = 0.0f;
}

__global__ void sage_count_edges(const int* __restrict__ dst,
                                 float* __restrict__ cnt, int E) {
    int e = blockIdx.x * blockDim.x + threadIdx.x;
    if (e < E) atomicAdd(&cnt[dst[e]], 1.0f);
}

__global__ void sage_inv_cnt(const float* __restrict__ cnt,
                             float* __restrict__ inv, int n) {
    int i = blockIdx.x * blockDim.x + threadIdx.x;
    if (i < n) inv[i] = 1.0f / fmaxf(cnt[i], 1.0f);
}

// ------------------------------------------------- edge scatter (mean accum)
// One wave32 per edge: lane l moves features [4l, 4l+4) as a coalesced float4
// gather from x[src], then 4 non-returning global_atomic_add_f32 into s[dst].

__global__ void sage_scatter(const float* __restrict__ x,
                             const int* __restrict__ src,
                             const int* __restrict__ dst,
                             float* __restrict__ s, int E) {
    int wave = (int)((blockIdx.x * (long long)blockDim.x + threadIdx.x) >> 5);
    int lane = threadIdx.x & 31;
    if (wave >= E) return;
    int sj = src[wave];
    int dj = dst[wave];
    const float4 v = *(const float4*)(x + (size_t)sj * FDIM + lane * 4);
    float* p = s + (size_t)dj * FDIM + lane * 4;
    atomicAdd(p + 0, v.x);
    atomicAdd(p + 1, v.y);
    atomicAdd(p + 2, v.z);
    atomicAdd(p + 3, v.w);
}

// ------------------------------------------------------ fused SAGE GEMM band
// out = act( (s .* inv) @ Wl + bl + h @ Wr )
// One wave per 16-row band (16x128 output). K=128 swept in 32 steps of
// v_wmma_f32_16x16x4_f32; per k-step the two A fragments (mean path, self
// path) are loaded once and reused by 8 column tiles -> 16 WMMAs per step.
//
// Fragment layouts (CDNA5 ISA 7.12.2, 32-bit, wave32):
//   A 16x4 : lane&15 = M; half = lane>=16 selects K{2,3}; vgpr0=K+0, vgpr1=K+1
//   B 4x16 : lane&15 = N; half selects K{2,3}; vgpr0=K+0, vgpr1=K+1
//   C/D    : lane&15 = N; vgpr r holds M=r (lanes 0-15) / M=r+8 (lanes 16-31)

__global__ void sage_gemm_band(const float* __restrict__ s,    // [N,128] sums
                               const float* __restrict__ inv,  // [N]
                               const float* __restrict__ h,    // [N,128] self
                               const float* __restrict__ Wl,   // [128,128]
                               const float* __restrict__ bias, // [128]
                               const float* __restrict__ Wr,   // [128,128]
                               float* __restrict__ out,        // [N,128]
                               int nrows, int relu) {
    int wid  = blockIdx.x * (blockDim.x >> 5) + (threadIdx.x >> 5); // row band
    int lane = threadIdx.x & 31;
    if (wid * 16 >= nrows) return;

    int half = lane >> 4;       // 0: K+{0,1}, 1: K+{2,3}
    int l    = lane & 15;
    int rowA = wid * 16 + l;    // M index for A fragments
    float invr = inv[rowA];

    v8f acc[8] = {};
    const float* ps = s + (size_t)rowA * FDIM;
    const float* ph = h + (size_t)rowA * FDIM;

    #pragma unroll 1
    for (int k = 0; k < FDIM; k += 4) {
        int ka = k + 2 * half;
        v2f am, ah;
        am.x = ps[ka]     * invr;
        am.y = ps[ka + 1] * invr;
        ah.x = ph[ka];
        ah.y = ph[ka + 1];
        const float* pwl = Wl + (size_t)ka * FDIM + l;
        const float* pwr = Wr + (size_t)ka * FDIM + l;
        #pragma unroll
        for (int tn = 0; tn < 8; ++tn) {
            v2f bwl, bwr;
            bwl.x = pwl[tn * 16];
            bwl.y = pwl[tn * 16 + FDIM];
            bwr.x = pwr[tn * 16];
            bwr.y = pwr[tn * 16 + FDIM];
            acc[tn] = __builtin_amdgcn_wmma_f32_16x16x4_f32(
                false, am, false, bwl, (short)0, acc[tn], false, false);
            acc[tn] = __builtin_amdgcn_wmma_f32_16x16x4_f32(
                false, ah, false, bwr, (short)0, acc[tn], false, false);
        }
    }

    #pragma unroll
    for (int tn = 0; tn < 8; ++tn) {
        float bv = bias[tn * 16 + l];
        #pragma unroll
        for (int r = 0; r < 8; ++r) {
            float v = acc[tn][r] + bv;
            if (relu) v = fmaxf(v, 0.0f);
            int row = wid * 16 + r + 8 * half;
            out[(size_t)row * FDIM + tn * 16 + l] = v;
        }
    }
}

// ----------------------------------------------------------------- launcher

static void run_layer(const float* in_feat, const int* src, const int* dst,
                      const float* inv, const float* Wl, const float* bl,
                      const float* Wr, float* s, float* out_feat,
                      int E, int relu, hipStream_t stream) {
    long long nElem = (long long)NNODES * FDIM;
    sage_zero_f32<<<(int)((nElem + 255) / 256), 256, 0, stream>>>(s, nElem);

    long long scatterThreads = 32LL * E;
    sage_scatter<<<(int)((scatterThreads + 255) / 256), 256, 0, stream>>>(
        in_feat, src, dst, s, E);

    int bands  = NNODES / 16;                 // 6250 row bands, 1 wave each
    int blocks = (bands + 7) / 8;             // 8 waves (256 thr) per block
    sage_gemm_band<<<blocks, 256, 0, stream>>>(s, inv, in_feat, Wl, bl, Wr,
                                               out_feat, NNODES, relu);
}

extern "C" void kernel_launch(void* const* d_in, const int* in_sizes, int n_in,
                              void* d_out, int out_size, void* d_ws, size_t ws_size,
                              hipStream_t stream) {
    const float* x   = (const float*)d_in[0];
    const int*   ei  = (const int*)d_in[1];
    const float* Wl0 = (const float*)d_in[2];
    const float* bl0 = (const float*)d_in[3];
    const float* Wr0 = (const float*)d_in[4];
    const float* Wl1 = (const float*)d_in[5];
    const float* bl1 = (const float*)d_in[6];
    const float* Wr1 = (const float*)d_in[7];
    const float* Wl2 = (const float*)d_in[8];
    const float* bl2 = (const float*)d_in[9];
    const float* Wr2 = (const float*)d_in[10];

    int E = in_sizes[1] / 2;            // edge_index is [2, E]
    const int* src = ei;
    const int* dst = ei + E;

    // Workspace layout: s | hA | hB | cnt | inv  (~155 MB)
    size_t feat = (size_t)NNODES * FDIM;
    float* s   = (float*)d_ws;
    float* hA  = s + feat;
    float* hB  = hA + feat;
    float* cnt = hB + feat;
    float* inv = cnt + NNODES;

    // Degree counts (shared across all three layers).
    sage_zero_f32<<<(NNODES + 255) / 256, 256, 0, stream>>>(cnt, NNODES);
    sage_count_edges<<<(E + 255) / 256, 256, 0, stream>>>(dst, cnt, E);
    sage_inv_cnt<<<(NNODES + 255) / 256, 256, 0, stream>>>(cnt, inv, NNODES);

    run_layer(x,  src, dst, inv, Wl0, bl0, Wr0, s, hA,            E, 1, stream);
    run_layer(hA, src, dst, inv, Wl1, bl1, Wr1, s, hB,            E, 1, stream);
    run_layer(hB, src, dst, inv, Wl2, bl2, Wr2, s, (float*)d_out, E, 0, stream);
}